// MLPAttn_66812511256676
// MI455X (gfx1250) — compile-verified
//
#include <hip/hip_runtime.h>

typedef __attribute__((ext_vector_type(2))) float v2f;
typedef __attribute__((ext_vector_type(8))) float v8f;

#define HDIM 256
#define NROWS 1024   // B * 512

// ---------------------------------------------------------------------------
// Kernel 1: both projections via V_WMMA_F32_16X16X4_F32 (one wave = one
// 16x16 output tile, K-loop of 64 wmma ops).
//   dp[(b*512+q)*256 + d] = sum_h dec[b,q,h] * W_mlp[d, 256+h]  + b_mlp[d]
//   ep[(b*512+e)*256 + d] = sum_h enc[b,e,h] * W_mlp[d, h]
// A-frag (16x4 f32): lanes 0-15 hold M=lane, VGPR0/1 = K=0/1; lanes 16-31
// hold K=2/3 -> per-lane contiguous float2 from the row-major source.
// B-frag (4x16 f32): lanes 0-15 N=lane with K=0/1 in VGPR0/1; lanes 16-31
// K=2/3 -> per-lane contiguous float2 from row-major W (W^T access).
// ---------------------------------------------------------------------------
__global__ __launch_bounds__(256) void proj_wmma(
    const float* __restrict__ dec, const float* __restrict__ enc,
    const float* __restrict__ W,   const float* __restrict__ bmlp,
    float* __restrict__ dp, float* __restrict__ ep)
{
  const int lane = threadIdx.x & 31;
  const int wv   = threadIdx.x >> 5;
  const int tile = blockIdx.x * 8 + wv;      // 0..2047 (uniform per wave)
  const bool isDec = tile < 1024;
  const int tt      = tile & 1023;
  const int rowTile = tt >> 4;               // 0..63
  const int colTile = tt & 15;               // 0..15
  const int row0 = rowTile << 4;
  const int col0 = colTile << 4;

  const float* src  = isDec ? dec : enc;
  const int    woff = isDec ? HDIM : 0;      // Wd lives at columns [256,512)
  float*       dst  = isDec ? dp : ep;

  const int laneLo = lane & 15;
  const int laneHi = lane >> 4;

  const float* aPtr = src + (size_t)(row0 + laneLo) * HDIM + 2 * laneHi;
  const float* bPtr = W   + (size_t)(col0 + laneLo) * (2 * HDIM) + woff + 2 * laneHi;

  v8f acc = {};
  for (int k = 0; k < HDIM; k += 4) {
    v2f a = *(const v2f*)(aPtr + k);
    v2f b = *(const v2f*)(bPtr + k);
    acc = __builtin_amdgcn_wmma_f32_16x16x4_f32(
        /*neg_a=*/false, a, /*neg_b=*/false, b,
        /*c_mod=*/(short)0, acc, /*reuse_a=*/false, /*reuse_b=*/false);
  }

  // C/D layout: VGPR v -> M = v + 8*laneHi, N = laneLo. Fold b_mlp into dp.
  const float bias = isDec ? bmlp[col0 + laneLo] : 0.0f;
  #pragma unroll
  for (int v = 0; v < 8; ++v) {
    const int m = v + 8 * laneHi;
    dst[(size_t)(row0 + m) * HDIM + col0 + laneLo] = acc[v] + bias;
  }
}

// ---------------------------------------------------------------------------
// Kernel 2: attn[b,q,e] = sum_h relu(dp[q,h] + ep[e,h]) * w[h] + b_out
// (bias already folded into dp). 256 threads -> 64x64 output tile, 4x4
// register blocking per thread, H streamed through LDS in 64-wide chunks,
// TWO h values consumed per step via ds_load_b64 (halves DS issue pressure:
// 9 DS instructions per 96 VALU ops).
// LDS row stride 66 floats = 264 B: rows stay 8B-aligned for b64 access;
// e-reads (16 distinct rows, stride 16) hit banks (2*tx+h) mod 64 ->
// 16 distinct even banks + adjacent odd bank per b64, conflict-free;
// q-reads are wave broadcasts.
// ---------------------------------------------------------------------------
#define HC  64
#define PAD 66

__global__ __launch_bounds__(256) void attn_combine(
    const float* __restrict__ dp, const float* __restrict__ ep,
    const float* __restrict__ wout, const float* __restrict__ bout,
    float* __restrict__ out)
{
  __shared__ __align__(16) float dpS[64][PAD];
  __shared__ __align__(16) float epS[64][PAD];
  __shared__ __align__(16) float wS[HDIM];

  const int tid = threadIdx.x;
  const int b   = blockIdx.x >> 6;
  const int qt  = (blockIdx.x >> 3) & 7;
  const int et  = blockIdx.x & 7;
  const int q0  = qt * 64, e0 = et * 64;
  const int tx  = tid & 15, ty = tid >> 4;

  wS[tid] = wout[tid];   // tid < 256 == HDIM

  const int r = tid >> 5;        // 0..7   (row group for cooperative loads)
  const int c = (tid & 31) * 2;  // 0..62  (even -> 8B-aligned v2f)

  float acc[4][4] = {};

  for (int hc = 0; hc < HDIM; hc += HC) {
    __syncthreads();   // protect LDS from previous iteration's readers
    #pragma unroll
    for (int i = 0; i < 8; ++i) {
      const int row = r + 8 * i;
      *(v2f*)&dpS[row][c] =
          *(const v2f*)(dp + (size_t)(b * 512 + q0 + row) * HDIM + hc + c);
      *(v2f*)&epS[row][c] =
          *(const v2f*)(ep + (size_t)(b * 512 + e0 + row) * HDIM + hc + c);
    }
    __syncthreads();

    for (int h = 0; h < HC; h += 2) {
      const v2f wv = *(const v2f*)&wS[hc + h];
      v2f dq[4], ee[4];
      #pragma unroll
      for (int i = 0; i < 4; ++i) dq[i] = *(const v2f*)&dpS[ty + 16 * i][h];
      #pragma unroll
      for (int j = 0; j < 4; ++j) ee[j] = *(const v2f*)&epS[tx + 16 * j][h];
      #pragma unroll
      for (int i = 0; i < 4; ++i)
        #pragma unroll
        for (int j = 0; j < 4; ++j) {
          acc[i][j] += fmaxf(dq[i].x + ee[j].x, 0.0f) * wv.x;
          acc[i][j] += fmaxf(dq[i].y + ee[j].y, 0.0f) * wv.y;
        }
    }
  }

  const float bo = bout[0];
  #pragma unroll
  for (int i = 0; i < 4; ++i)
    #pragma unroll
    for (int j = 0; j < 4; ++j)
      out[(size_t)(b * 512 + q0 + ty + 16 * i) * 512 + e0 + tx + 16 * j] =
          acc[i][j] + bo;
}

// ---------------------------------------------------------------------------
extern "C" void kernel_launch(void* const* d_in, const int* in_sizes, int n_in,
                              void* d_out, int out_size, void* d_ws, size_t ws_size,
                              hipStream_t stream) {
  const float* dec = (const float*)d_in[0];  // (2,512,256)
  const float* enc = (const float*)d_in[1];  // (2,512,256)
  const float* W   = (const float*)d_in[2];  // (256,512)
  const float* bm  = (const float*)d_in[3];  // (256,)
  const float* wo  = (const float*)d_in[4];  // (1,256)
  const float* bo  = (const float*)d_in[5];  // (1,)
  float* out = (float*)d_out;                // (2,512,512)

  float* dp = (float*)d_ws;                  // 1024*256 f32 (bias folded in)
  float* ep = dp + (size_t)NROWS * HDIM;     // 1024*256 f32

  // 2048 WMMA tiles total (dec + enc projections), 8 waves per block.
  proj_wmma<<<256, 256, 0, stream>>>(dec, enc, W, bm, dp, ep);
  // 2 * 8 * 8 = 128 output tiles of 64x64.
  attn_combine<<<128, 256, 0, stream>>>(dp, ep, wo, bo, out);
}